// BinsChamferLoss_1382979469676
// MI455X (gfx1250) — compile-verified
//
#include <hip/hip_runtime.h>

// CDNA5 / gfx1250 ext-vector types for WMMA operands
typedef float v2f __attribute__((ext_vector_type(2)));
typedef float v8f __attribute__((ext_vector_type(8)));

#define NB 4        // batches
#define NP 256      // bin centers per batch
#define NL 76800    // depth points per batch (240*320)
#define CHUNK 1280  // points per block
#define NCH (NL/CHUNK) // 60 point-chunks (blocks) per batch
#define WPTS 160    // points per wave   (8 waves * 160 = CHUNK)
#define NTILE 10    // 16-point tiles per wave
#define THRESH 1e-3f
#define SENTINEL 1e18f  // replaces invalid depths: finite, never wins a min

// workspace layout (uint words)
#define WS_MINX 0                   // NB*NP: per-center min diff^2 (float bits)
#define WS_SUMY (NB*NP)             // NB*NCH: per-block cham_y partial sums (float bits)
#define WS_CNT  (NB*NP + NB*NCH)    // NB*NCH: per-block valid-point counts
#define WS_TOTAL (NB*NP + 2*NB*NCH) // 1504 words = 6016 bytes

__global__ void init_ws(unsigned* __restrict__ ws) {
  // Only the atomicMin targets need re-init each launch; partial-sum slots
  // are overwritten unconditionally by exactly one block each.
  for (int k = threadIdx.x; k < NB*NP; k += blockDim.x)
    ws[k] = 0x7F800000u; // +inf bits
}

__device__ __forceinline__ unsigned umin2(unsigned a, unsigned b) {
  return a < b ? a : b;  // v_min_u32; u32 order == float order for >= 0
}

// One block = one (batch, 1280-point chunk). 8 wave32s; each wave sweeps all
// 256 centers against its 160 points using V_WMMA_F32_16X16X4_F32 to form
// exact (bc[m] - tp[j]) 16x16 tiles:  A = [bc, 1], B = [ones; -tp].
// Per-center minima: butterfly-fold 8 regs x 16 lanes -> 16 values in
// 8 shuffles, one conflict-free ds_min_u32, then one global atomicMin per
// center per block. u32 min is exact & order-independent and cham_y partials
// are single-writer slots => whole pipeline is bit-deterministic.
__global__ void __launch_bounds__(256) cham_main(const float* __restrict__ bins,
                                                 const float* __restrict__ tdm,
                                                 unsigned* __restrict__ ws) {
  __shared__ float    bc[NP];
  __shared__ unsigned smin[NP];   // block-level per-center minima (float bits)
  __shared__ float    swsum[8];   // per-wave cham_y sums
  __shared__ int      swcnt[8];   // per-wave valid counts
  const int n   = blockIdx.y;
  const int tid = threadIdx.x;

  // bin centers -> LDS; init block minima
  bc[tid]   = 0.5f * (bins[n*(NP+1) + tid] + bins[n*(NP+1) + tid + 1]);
  smin[tid] = 0x7F800000u;
  __syncthreads();

  const int  lane = tid & 31;
  const int  wv   = tid >> 5;
  const bool lo   = lane < 16;          // WMMA low half (K0/K1 rows, M=v rows)
  const float one16 = lo ? 1.0f : 0.0f; // "1" in K0/K1 slots, 0 in K2/K3 slots
  const int  pbase = n*NL + blockIdx.x*CHUNK + wv*WPTS;

  // Stage this wave's 10 point-tiles in registers as B.y (= -tp in row K1,
  // lanes 0-15; 0 in row K3, lanes 16-31). Invalid depths -> sentinel.
  float negtp[NTILE];
  float miny[NTILE];
  unsigned vbits = 0;
  #pragma unroll
  for (int t = 0; t < NTILE; ++t) {
    float v = tdm[pbase + t*16 + (lane & 15)];
    bool valid = (v >= THRESH);
    if (valid) vbits |= (1u << t);
    float vm = valid ? v : SENTINEL;
    negtp[t] = lo ? -vm : 0.0f;
    miny[t]  = __builtin_inff();
  }

  // Software-pipelined A-operand: prefetch next group's bin centers so the
  // ds_load latency is hidden behind the current group's WMMA+VALU body.
  float bcl = bc[lane & 15];            // group 0
  for (int g = 0; g < 16; ++g) {        // 16 center-groups of 16 centers
    int gn = (g + 1) & 15;              // wraps to 0 on last iter (harmless)
    float bcl_next = bc[gn*16 + (lane & 15)];

    v2f A;
    A.x = lo ? bcl : 0.0f;              // K0 col = bc[m]; K2 col = 0
    A.y = one16;                        // K1 col = 1;     K3 col = 0

    float minx[8];
    #pragma unroll
    for (int v = 0; v < 8; ++v) minx[v] = __builtin_inff();

    #pragma unroll
    for (int t = 0; t < NTILE; ++t) {
      v2f B;
      B.x = one16;      // row K0 = ones (lanes 0-15); row K2 = 0
      B.y = negtp[t];   // row K1 = -tp  (lanes 0-15); row K3 = 0
      v8f C = {};
      // D[m][j] = bc[m]*1 + 1*(-tp[j]) = bc[m] - tp[j]   (exact in f32)
      v8f D = __builtin_amdgcn_wmma_f32_16x16x4_f32(
                  false, A, false, B, (short)0, C, false, false);
      float m = __builtin_inff();
      #pragma unroll
      for (int v = 0; v < 8; ++v) {
        float d = D[v] * D[v];               // squared distance
        minx[v] = fminf(minx[v], d);         // per-center running min
        m = fminf(m, d);                     // per-point partial min over M
      }
      miny[t] = fminf(miny[t], m);
    }

    // Butterfly fold: 8 regs x 16 lanes -> lane L holds min for v = L&7,
    // reduced over its 16-lane half. u32 domain: no NaN canonicalization.
    unsigned r[8];
    #pragma unroll
    for (int v = 0; v < 8; ++v) r[v] = __float_as_uint(minx[v]);

    unsigned s4[4];
    #pragma unroll
    for (int v = 0; v < 4; ++v) {           // fold by lane bit0
      unsigned keep = (lane & 1) ? r[2*v+1] : r[2*v];
      unsigned send = (lane & 1) ? r[2*v]   : r[2*v+1];
      s4[v] = umin2(keep, __shfl_xor(send, 1, 32));
    }
    unsigned s2[2];
    #pragma unroll
    for (int v = 0; v < 2; ++v) {           // fold by lane bit1
      unsigned keep = (lane & 2) ? s4[2*v+1] : s4[2*v];
      unsigned send = (lane & 2) ? s4[2*v]   : s4[2*v+1];
      s2[v] = umin2(keep, __shfl_xor(send, 2, 32));
    }
    {                                        // fold by lane bit2
      unsigned keep = (lane & 4) ? s2[1] : s2[0];
      unsigned send = (lane & 4) ? s2[0] : s2[1];
      unsigned u = umin2(keep, __shfl_xor(send, 4, 32));
      u = umin2(u, __shfl_xor(u, 8, 32));    // cover remaining lane pairs
      // lane L: center g*16 + (L&7) + (L<16 ? 0 : 8); lanes L and L^8 dup.
      if ((lane & 8) == 0)                   // 16 lanes, 16 distinct banks
        atomicMin(&smin[g*16 + (lane & 7) + (lo ? 0 : 8)], u);
    }

    bcl = bcl_next;
  }

  // cham_y: combine the two M-halves per point, mask invalid, reduce wave-wide
  // with a fixed-order shuffle tree (deterministic).
  float sumy = 0.0f;
  int   cnty = 0;
  #pragma unroll
  for (int t = 0; t < NTILE; ++t) {
    float m = fminf(miny[t], __shfl_xor(miny[t], 16, 32));
    if (lo && ((vbits >> t) & 1u)) { sumy += m; cnty += 1; }
  }
  #pragma unroll
  for (int off = 16; off; off >>= 1) {
    sumy += __shfl_xor(sumy, off, 32);
    cnty += __shfl_xor(cnty, off, 32);
  }
  if (lane == 0) { swsum[wv] = sumy; swcnt[wv] = cnty; }
  __syncthreads();

  // One global atomicMin per center per block (order-independent u32 min).
  atomicMin(&ws[WS_MINX + n*NP + tid], smin[tid]);

  // Single-writer per-block cham_y partials, folded in fixed order.
  if (tid == 0) {
    float s = 0.0f; int c = 0;
    #pragma unroll
    for (int w = 0; w < 8; ++w) { s += swsum[w]; c += swcnt[w]; }
    ws[WS_SUMY + n*NCH + blockIdx.x] = __float_as_uint(s);
    ws[WS_CNT  + n*NCH + blockIdx.x] = (unsigned)c;
  }
}

__global__ void finalize(const unsigned* __restrict__ ws, float* __restrict__ out) {
  __shared__ float sx[NP];
  int t = threadIdx.x;
  float total = 0.0f;
  for (int n = 0; n < NB; ++n) {
    sx[t] = __uint_as_float(ws[WS_MINX + n*NP + t]);
    __syncthreads();
    for (int s = NP/2; s > 0; s >>= 1) {
      if (t < s) sx[t] += sx[t + s];
      __syncthreads();
    }
    if (t == 0) {
      float cham_x = sx[0] * (1.0f / NP);
      float sum_y = 0.0f; unsigned cnt = 0;
      for (int c = 0; c < NCH; ++c) {       // fixed order -> deterministic
        sum_y += __uint_as_float(ws[WS_SUMY + n*NCH + c]);
        cnt   += ws[WS_CNT + n*NCH + c];
      }
      total += cham_x + sum_y / (float)cnt;
    }
    __syncthreads();
  }
  if (t == 0) out[0] = total * (1.0f / NB);
}

extern "C" void kernel_launch(void* const* d_in, const int* in_sizes, int n_in,
                              void* d_out, int out_size, void* d_ws, size_t ws_size,
                              hipStream_t stream) {
  const float* bins = (const float*)d_in[0];   // [4, 257]
  const float* tdm  = (const float*)d_in[1];   // [4, 240, 320]
  unsigned* ws  = (unsigned*)d_ws;             // needs WS_TOTAL*4 = 6016 bytes
  float*    out = (float*)d_out;

  init_ws<<<1, 256, 0, stream>>>(ws);
  cham_main<<<dim3(NCH, NB), 256, 0, stream>>>(bins, tdm, ws);
  finalize<<<1, NP, 0, stream>>>(ws, out);
}